// Head_36670430773709
// MI455X (gfx1250) — compile-verified
//
#include <hip/hip_runtime.h>
#include <math.h>
#include <stdint.h>

#define BATCH    512
#define SEQ      142
#define TP       144          // seq padded to 9 tiles of 16
#define CDIM     512
#define HS       64
#define KCHUNK   128          // K split into 4 chunks of 128
#define NCHUNK   (CDIM / KCHUNK)
#define NTHREADS 256

typedef __attribute__((ext_vector_type(2))) float v2f;
typedef __attribute__((ext_vector_type(4))) float v4f;
typedef __attribute__((ext_vector_type(8))) float v8f;
typedef __attribute__((ext_vector_type(4))) unsigned int v4u;
typedef __attribute__((ext_vector_type(4))) int v4i;
typedef __attribute__((ext_vector_type(8))) int v8i;

// ---- TDM availability / arity detection -----------------------------------
#if __has_builtin(__builtin_amdgcn_tensor_load_to_lds)
  #define HAVE_TDM 1
  #if __has_include(<hip/amd_detail/amd_gfx1250_TDM.h>)
    #define HAVE_TDM6 1   // therock / clang-23: 6-arg form
  #endif
#else
  #define HAVE_TDM 0
#endif

// LDS layout (in floats)
#define OFF_Q      0                       // 9216: raw k -> later q = rot(rot(k))
#define OFF_K      (TP*HS)                 // 9216: k_rot
#define OFF_V      (2*TP*HS)               // 9216: v
#define OFF_STAGE0 (3*TP*HS)               // 18432: x-chunk stage buffer 0
#define OFF_STAGE1 (3*TP*HS + TP*KCHUNK)   // 18432: x-chunk stage buffer 1
#define OFF_S      (3*TP*HS)               // 20736: 144x144 scores (overlays stages)
#define SMEM_FLOATS (3*TP*HS + 2*TP*KCHUNK)
#define SMEM_BYTES  (SMEM_FLOATS * 4)

#if HAVE_TDM
// Issue one 2D TDM load: 142 rows x 128 f32 from global (row stride 512 f32)
// into LDS at lds_byte_off (contiguous 128-f32 rows). D# per ISA ch.8.
__device__ __forceinline__ void tdm_load_chunk(const float* gsrc,
                                               unsigned lds_byte_off) {
  unsigned long long ga = (unsigned long long)(uintptr_t)gsrc;
  v4u g0;
  g0.x = 0x1u;                                        // count=1, user mode
  g0.y = lds_byte_off;                                // lds_addr (bytes)
  g0.z = (unsigned)(ga & 0xFFFFFFFFu);                // global_addr[31:0]
  g0.w = (unsigned)((ga >> 32) & 0x1FFFFFFu)          // global_addr[56:32]
       | (2u << 30);                                  // type = 2 ("image")
  v8i g1;
  g1[0] = 0x20000;                  // workgroup_mask=0, data_size=2 (4 bytes)
  g1[1] = (int)(512u << 16);        // tensor_dim0 = 512 (low16 at [63:48])
  g1[2] = (int)(142u << 16);        // tensor_dim0 hi=0; tensor_dim1 = 142
  g1[3] = (int)(128u << 16);        // tensor_dim1 hi=0; tile_dim0 = 128
  g1[4] = 142;                      // tile_dim1 = 142; tile_dim2 = 0
  g1[5] = 512;                      // tensor_dim0_stride = 512 (elements)
  g1[6] = 0;                        // stride hi = 0; tensor_dim1_stride = 0
  g1[7] = 0;
  v4i z4 = {};
#if defined(HAVE_TDM6)
  v8i z8 = {};
  __builtin_amdgcn_tensor_load_to_lds(g0, g1, z4, z4, z8, 0);
#else
  __builtin_amdgcn_tensor_load_to_lds(g0, g1, z4, z4, 0);
#endif
}
#endif

__global__ __launch_bounds__(NTHREADS, 1)
void rope_attn_head_kernel(const float* __restrict__ x,
                           const float* __restrict__ Wk,
                           const float* __restrict__ Wv,
                           float* __restrict__ out) {
  extern __shared__ float smem[];
  float* bufQ   = smem + OFF_Q;        // raw k -> q (double-rotated key)
  float* bufK   = smem + OFF_K;        // k_rot
  float* bufV   = smem + OFF_V;
  float* stage0 = smem + OFF_STAGE0;   // 144 x 128 x-chunk (double-buffered)
  float* stage1 = smem + OFF_STAGE1;
  float* ss     = smem + OFF_S;        // 144 x 144 scores (reuses stage space)

  const int tid  = threadIdx.x;
  const int wave = tid >> 5;
  const int lane = tid & 31;
  const int b    = blockIdx.x;

  const int m  = lane & 15;            // fragment row/col within half-wave
  const int hk = lane >> 4;            // half-wave select (0/1)
  const float scale = 0.044194173824159216f;   // 512^-0.5
  const float NEG_INF = -__builtin_inff();

  // ---------------- Phase 1: k = x@Wk, v = x@Wv ----------------------------
  // Zero the pad rows (142,143) of both stage buffers (TDM never writes them).
  {
    const int row = 142 + (tid >> 7), c = tid & 127;
    stage0[row * KCHUNK + c] = 0.0f;
    stage1[row * KCHUNK + c] = 0.0f;
  }

#if HAVE_TDM
  if (wave == 0)
    tdm_load_chunk(x + (size_t)b * SEQ * CDIM, OFF_STAGE0 * 4);
#endif

  const float* W  = (wave < 4) ? Wk : Wv;   // waves 0-3: k, waves 4-7: v
  const int ct    = wave & 3;
  const int col   = ct * 16 + m;

  v8f acc[9];
  #pragma unroll
  for (int rt = 0; rt < 9; ++rt) acc[rt] = (v8f){};

  for (int chunk = 0; chunk < NCHUNK; ++chunk) {
    float* cur = (chunk & 1) ? stage1 : stage0;
#if HAVE_TDM
    if (wave == 0) __builtin_amdgcn_s_wait_tensorcnt(0);   // chunk's DMA done
    __syncthreads();                                       // publish to all waves
    if (wave == 0 && chunk < NCHUNK - 1)                   // prefetch next chunk
      tdm_load_chunk(x + (size_t)b * SEQ * CDIM + (chunk + 1) * KCHUNK,
                     ((chunk & 1) ? OFF_STAGE0 : OFF_STAGE1) * 4);
#else
    for (int f = tid; f < SEQ * (KCHUNK / 4); f += NTHREADS) {
      const int row = f >> 5, c4 = (f & 31) * 4;
      *(v4f*)(cur + row * KCHUNK + c4) =
          *(const v4f*)(x + ((size_t)b * SEQ + row) * CDIM + chunk * KCHUNK + c4);
    }
    __syncthreads();
#endif

    // B fragments for this K-chunk live in registers; reused across 9 row-tiles.
    v2f breg[32];
    #pragma unroll
    for (int i = 0; i < 32; ++i) {
      const int koff = chunk * KCHUNK + i * 4 + (hk << 1);
      breg[i].x = W[(size_t)koff * HS + col];
      breg[i].y = W[(size_t)(koff + 1) * HS + col];
    }

    #pragma unroll
    for (int rt = 0; rt < 9; ++rt) {
      const float* arow = cur + (rt * 16 + m) * KCHUNK + (hk << 1);
      #pragma unroll
      for (int i = 0; i < 32; ++i) {
        v2f a = *(const v2f*)(arow + i * 4);
        acc[rt] = __builtin_amdgcn_wmma_f32_16x16x4_f32(false, a, false, breg[i],
                                                        (short)0, acc[rt],
                                                        false, false);
      }
    }
    __syncthreads();   // all waves done reading cur before it is overwritten
  }

  // Write projection tiles to LDS (raw k -> bufQ, v -> bufV).
  {
    float* dst = (wave < 4) ? bufQ : bufV;
    #pragma unroll
    for (int rt = 0; rt < 9; ++rt) {
      #pragma unroll
      for (int r = 0; r < 8; ++r) {
        const int t = rt * 16 + r + (hk << 3);
        dst[t * HS + col] = acc[rt][r];
      }
    }
  }
  __syncthreads();

  // ---------------- Phase 1b: RoPE (k_rot = rot(k); q = rot(k_rot)) --------
  for (int e = tid; e < TP * HS; e += NTHREADS) {
    const int t = e >> 6, h = e & 63;
    const float theta = expf(-(float)(h & 62) * (9.2103403720f / 64.0f));
    const float ang = (float)t * theta;
    const float cs = cosf(ang), sn = sinf(ang);
    const float raw = bufQ[e];
    const float rh  = (h & 1) ? bufQ[e - 1] : -bufQ[e + 1];
    bufK[e] = raw * cs + rh * sn;                  // k_rot
  }
  __syncthreads();
  for (int e = tid; e < TP * HS; e += NTHREADS) {
    const int t = e >> 6, h = e & 63;
    const float theta = expf(-(float)(h & 62) * (9.2103403720f / 64.0f));
    const float ang = (float)t * theta;
    const float cs = cosf(ang), sn = sinf(ang);
    const float raw = bufK[e];
    const float rh  = (h & 1) ? bufK[e - 1] : -bufK[e + 1];
    bufQ[e] = raw * cs + rh * sn;                  // q = rot(k_rot)
  }
  __syncthreads();

  // ---------------- Phase 2: scores = q @ k_rot^T, scaled + causal mask ----
  for (int tile = wave; tile < 81; tile += 8) {
    const int rt = tile / 9, cc = tile % 9;
    const float* qrow = bufQ + (rt * 16 + m) * HS;
    const float* krow = bufK + (cc * 16 + m) * HS;
    v8f sacc = {};
    #pragma unroll
    for (int kb = 0; kb < HS; kb += 4) {
      const int koff = kb + (hk << 1);
      v2f a  = *(const v2f*)(qrow + koff);
      v2f bb = *(const v2f*)(krow + koff);
      sacc = __builtin_amdgcn_wmma_f32_16x16x4_f32(false, a, false, bb,
                                                   (short)0, sacc, false, false);
    }
    #pragma unroll
    for (int r = 0; r < 8; ++r) {
      const int t = rt * 16 + r + (hk << 3);
      const int s = cc * 16 + m;
      float v = sacc[r] * scale;
      if (s > t || s >= SEQ) v = NEG_INF;
      ss[t * TP + s] = v;
    }
  }
  __syncthreads();

  // ---------------- Softmax: one thread per row ----------------------------
  if (tid < TP) {
    float* row = ss + tid * TP;
    if (tid >= SEQ) {
      for (int s = 0; s < TP; ++s) row[s] = 0.0f;
    } else {
      float mx = NEG_INF;
      for (int s = 0; s < TP; ++s) mx = fmaxf(mx, row[s]);
      float sum = 0.0f;
      for (int s = 0; s < TP; ++s) { float e = expf(row[s] - mx); row[s] = e; sum += e; }
      const float inv = 1.0f / sum;
      for (int s = 0; s < TP; ++s) row[s] *= inv;
    }
  }
  __syncthreads();

  // ---------------- Phase 3: out = P @ v -----------------------------------
  for (int tile = wave; tile < 36; tile += 8) {
    const int rt = tile >> 2, cc = tile & 3;
    const int oc = cc * 16 + m;
    const float* prow = ss + (rt * 16 + m) * TP;
    v8f oacc = {};
    #pragma unroll
    for (int kb = 0; kb < TP; kb += 4) {
      const int koff = kb + (hk << 1);
      v2f a = *(const v2f*)(prow + koff);
      v2f bb;
      bb.x = bufV[koff * HS + oc];
      bb.y = bufV[(koff + 1) * HS + oc];
      oacc = __builtin_amdgcn_wmma_f32_16x16x4_f32(false, a, false, bb,
                                                   (short)0, oacc, false, false);
    }
    #pragma unroll
    for (int r = 0; r < 8; ++r) {
      const int t = rt * 16 + r + (hk << 3);
      if (t < SEQ) out[((size_t)b * SEQ + t) * HS + oc] = oacc[r];
    }
  }
}

extern "C" void kernel_launch(void* const* d_in, const int* in_sizes, int n_in,
                              void* d_out, int out_size, void* d_ws, size_t ws_size,
                              hipStream_t stream) {
  const float* x  = (const float*)d_in[0];
  const float* Wk = (const float*)d_in[1];
  // d_in[2] = Wq is dead in the reference (q_unused); never touched.
  const float* Wv = (const float*)d_in[3];
  float* out = (float*)d_out;

  (void)hipFuncSetAttribute((const void*)rope_attn_head_kernel,
                            hipFuncAttributeMaxDynamicSharedMemorySize, SMEM_BYTES);
  rope_attn_head_kernel<<<dim3(BATCH), dim3(NTHREADS), SMEM_BYTES, stream>>>(x, Wk, Wv, out);
}